// Structure2Vec_41162966565590
// MI455X (gfx1250) — compile-verified
//
#include <hip/hip_runtime.h>

typedef float v2f __attribute__((ext_vector_type(2)));
typedef float v8f __attribute__((ext_vector_type(8)));

#define HDIM 128
#define EPSV 1e-5f

// ---------------- utility ----------------
__global__ void k_zero(float* __restrict__ p, long n) {
  long i = (long)blockIdx.x * blockDim.x + threadIdx.x;
  long st = (long)gridDim.x * blockDim.x;
  for (; i < n; i += st) p[i] = 0.0f;
}
__global__ void k_zeroi(int* __restrict__ p, long n) {
  long i = (long)blockIdx.x * blockDim.x + threadIdx.x;
  long st = (long)gridDim.x * blockDim.x;
  for (; i < n; i += st) p[i] = 0;
}

// ---------------- CSR build (once per launch) ----------------
__global__ void k_count(const int* __restrict__ dst, int* __restrict__ cnt, int E) {
  int e = blockIdx.x * blockDim.x + threadIdx.x;
  if (e < E) atomicAdd(&cnt[dst[e]], 1);
}

// single-block chunked exclusive scan: rowptr[0..N], rowptr[N] = E
__global__ __launch_bounds__(1024)
void k_scan(const int* __restrict__ cnt, int* __restrict__ rowptr, int N) {
  __shared__ int sh[1024];
  __shared__ int soff;
  if (threadIdx.x == 0) soff = 0;
  __syncthreads();
  for (int base = 0; base < N; base += 1024) {
    int i = base + threadIdx.x;
    int v = (i < N) ? cnt[i] : 0;
    sh[threadIdx.x] = v;
    __syncthreads();
    for (int d = 1; d < 1024; d <<= 1) {       // inclusive Hillis-Steele
      int t = (threadIdx.x >= d) ? sh[threadIdx.x - d] : 0;
      __syncthreads();
      sh[threadIdx.x] += t;
      __syncthreads();
    }
    int incl = sh[threadIdx.x];
    if (i < N) rowptr[i] = soff + incl - v;    // exclusive
    __syncthreads();
    if (threadIdx.x == 1023) soff += sh[1023];
    __syncthreads();
  }
  if (threadIdx.x == 0) rowptr[N] = soff;
}

__global__ void k_fill(const int* __restrict__ dst, const int* __restrict__ rowptr,
                       int* __restrict__ fill, int* __restrict__ eid, int E) {
  int e = blockIdx.x * blockDim.x + threadIdx.x;
  if (e >= E) return;
  int d = dst[e];
  int pos = rowptr[d] + atomicAdd(&fill[d], 1);
  eid[pos] = e;
}

// ws_agg[n,0:16] = sum over incoming edges of w[e,0:16]  (pure gather, no atomics)
__global__ __launch_bounds__(256)
void k_ws_agg(const float* __restrict__ w, const int* __restrict__ rowptr,
              const int* __restrict__ eid, float* __restrict__ ws_agg, int N) {
  int wave = threadIdx.x >> 5, lane = threadIdx.x & 31;
  int n = blockIdx.x * 8 + wave;
  if (n >= N || lane >= 16) return;
  int j0 = rowptr[n], j1 = rowptr[n + 1];
  float s = 0.0f;
  for (int j = j0; j < j1; ++j) s += w[(long)eid[j] * 16 + lane];
  ws_agg[(long)n * 16 + lane] = s;
}

// agg[n] = sum_{incoming e} h[src[e]]   (wave/node, float4/lane, register accum)
__global__ __launch_bounds__(256)
void k_aggregate(const float* __restrict__ h, const int* __restrict__ rowptr,
                 const int* __restrict__ eid, const int* __restrict__ src,
                 float* __restrict__ agg, int N) {
  int wave = threadIdx.x >> 5, lane = threadIdx.x & 31;
  int n = blockIdx.x * 8 + wave;
  if (n >= N) return;
  int j0 = rowptr[n], j1 = rowptr[n + 1];
  float ax = 0.f, ay = 0.f, az = 0.f, aw = 0.f;
  for (int j = j0; j < j1; ++j) {
    int s = src[eid[j]];
    const float4 v = *reinterpret_cast<const float4*>(h + (long)s * HDIM + lane * 4);
    ax += v.x; ay += v.y; az += v.z; aw += v.w;
  }
  float4 o; o.x = ax; o.y = ay; o.z = az; o.w = aw;
  *reinterpret_cast<float4*>(agg + (long)n * HDIM + lane * 4) = o;
}

// ---------------- weight pre-swizzle into WMMA fragment order ----------------
// Wf[kp][t][lane] = float4{ W[8kp+kb][col], W[8kp+kb+1][col], W[8kp+4+kb][col], W[8kp+4+kb+1][col] }
// with m = lane&15, kb = (lane>>4)*2, col = t*16+m  -> inner-loop B loads become coalesced b128
__global__ void k_swz(const float* __restrict__ W, float* __restrict__ Wf, int K) {
  int tid = blockIdx.x * blockDim.x + threadIdx.x;
  int total = K * 32;                 // (K/8) * 8tiles * 32lanes
  if (tid >= total) return;
  int l = tid & 31;
  int t = (tid >> 5) & 7;
  int kp = tid >> 8;
  int m = l & 15;
  int kb = (l >> 4) << 1;
  int col = t * 16 + m;
  int r0 = kp * 8 + kb;
  float4 o;
  o.x = W[(long)(r0 + 0) * HDIM + col];
  o.y = W[(long)(r0 + 1) * HDIM + col];
  o.z = W[(long)(r0 + 4) * HDIM + col];
  o.w = W[(long)(r0 + 5) * HDIM + col];
  *reinterpret_cast<float4*>(Wf + (long)tid * 4) = o;
}

// ---------------- fused WMMA GEMM ----------------
// Z = Amat[N,K1] @ W (Wf1 = swizzled) + b1
//     (+ ws_agg[N,16] @ Wb (Wfb swizzled) + deg[row]*bb[col])  if ws_agg != null
//     (+ addmat[row,col])                                      if addmat != null
// r = relu(Z) -> outR ; BN stats (sum, sumsq per column) -> gstats[2*H]
__global__ __launch_bounds__(256)
void k_gemm(const float* __restrict__ Amat, int K1, const float* __restrict__ Wf1,
            const float* __restrict__ b1,
            const float* __restrict__ ws_agg, const float* __restrict__ Wfb,
            const float* __restrict__ bb, const int* __restrict__ rowptr,
            const float* __restrict__ addmat,
            float* __restrict__ outR, float* __restrict__ gstats, int N) {
  __shared__ float slds[2 * HDIM];
  if (threadIdx.x < 2 * HDIM) slds[threadIdx.x] = 0.0f;
  __syncthreads();

  const int wave = threadIdx.x >> 5;   // wave32
  const int lane = threadIdx.x & 31;
  const int row0 = blockIdx.x * 128 + wave * 16;

  if (row0 < N) {
    const v8f vzero = {0.f, 0.f, 0.f, 0.f, 0.f, 0.f, 0.f, 0.f};
    v8f acc[8];
#pragma unroll
    for (int t = 0; t < 8; ++t) acc[t] = vzero;

    const int m  = lane & 15;
    const int kb = (lane >> 4) << 1;
    const long arow = row0 + m;

    // main GEMM: 8 K-values (2 wmma) per iteration
    for (int kp = 0; kp < (K1 >> 3); ++kp) {
      const float* ap = Amat + arow * K1 + kp * 8 + kb;
      v2f af0, af1;
      af0.x = ap[0]; af0.y = ap[1];
      af1.x = ap[4]; af1.y = ap[5];
      const float* bp = Wf1 + (long)kp * 1024 + (long)lane * 4;   // [kp][t][lane][4]
#pragma unroll
      for (int t = 0; t < 8; ++t) {
        const float4 bq = *reinterpret_cast<const float4*>(bp + t * 128);
        v2f bf0, bf1;
        bf0.x = bq.x; bf0.y = bq.y;
        bf1.x = bq.z; bf1.y = bq.w;
        acc[t] = __builtin_amdgcn_wmma_f32_16x16x4_f32(false, af0, false, bf0,
                                                       (short)0, acc[t], false, false);
        acc[t] = __builtin_amdgcn_wmma_f32_16x16x4_f32(false, af1, false, bf1,
                                                       (short)0, acc[t], false, false);
      }
    }

    // fused edge-GEMM: ws_agg[N,16] @ Wb[16,H]
    if (ws_agg) {
      for (int kp = 0; kp < 2; ++kp) {
        const float* ap = ws_agg + arow * 16 + kp * 8 + kb;
        v2f af0, af1;
        af0.x = ap[0]; af0.y = ap[1];
        af1.x = ap[4]; af1.y = ap[5];
        const float* bp = Wfb + (long)kp * 1024 + (long)lane * 4;
#pragma unroll
        for (int t = 0; t < 8; ++t) {
          const float4 bq = *reinterpret_cast<const float4*>(bp + t * 128);
          v2f bf0, bf1;
          bf0.x = bq.x; bf0.y = bq.y;
          bf1.x = bq.z; bf1.y = bq.w;
          acc[t] = __builtin_amdgcn_wmma_f32_16x16x4_f32(false, af0, false, bf0,
                                                         (short)0, acc[t], false, false);
          acc[t] = __builtin_amdgcn_wmma_f32_16x16x4_f32(false, af1, false, bf1,
                                                         (short)0, acc[t], false, false);
        }
      }
    }

    // epilogue
    const int rbase = row0 + ((lane >> 4) << 3);
    float degv[8];
#pragma unroll
    for (int i = 0; i < 8; ++i)
      degv[i] = ws_agg ? (float)(rowptr[rbase + i + 1] - rowptr[rbase + i]) : 0.0f;

#pragma unroll
    for (int t = 0; t < 8; ++t) {
      const int col = t * 16 + m;
      const float bias = b1[col];
      const float bbv = ws_agg ? bb[col] : 0.0f;
      float s = 0.0f, sq = 0.0f;
#pragma unroll
      for (int i = 0; i < 8; ++i) {
        const long rr = rbase + i;
        float z = acc[t][i] + bias + degv[i] * bbv;
        if (addmat) z += addmat[rr * HDIM + col];
        z = fmaxf(z, 0.0f);
        outR[rr * HDIM + col] = z;
        s += z;
        sq += z * z;
      }
      atomicAdd(&slds[col], s);            // ds_add_f32
      atomicAdd(&slds[HDIM + col], sq);
    }
  }
  __syncthreads();
  atomicAdd(&gstats[threadIdx.x], slds[threadIdx.x]);
}

// ---------------- BatchNorm ----------------
__global__ void k_bn_fin(const float* __restrict__ gstats, const float* __restrict__ g,
                         const float* __restrict__ be, float* __restrict__ ab, float invN) {
  int c = threadIdx.x;   // 128 threads, 1 block
  float mean = gstats[c] * invN;
  float var = gstats[HDIM + c] * invN - mean * mean;
  var = fmaxf(var, 0.0f);
  float a = g[c] * rsqrtf(var + EPSV);
  ab[c] = a;
  ab[HDIM + c] = be[c] - mean * a;
}

__global__ void k_bn_apply(const float* __restrict__ r, const float* __restrict__ ab,
                           float* __restrict__ out, int N) {
  int tid = blockIdx.x * blockDim.x + threadIdx.x;
  int total = N * (HDIM / 4);
  if (tid >= total) return;
  int c = (tid & (HDIM / 4 - 1)) << 2;
  const float4 v = *reinterpret_cast<const float4*>(r + (long)tid * 4);
  float4 o;
  o.x = v.x * ab[c + 0] + ab[HDIM + c + 0];
  o.y = v.y * ab[c + 1] + ab[HDIM + c + 1];
  o.z = v.z * ab[c + 2] + ab[HDIM + c + 2];
  o.w = v.w * ab[c + 3] + ab[HDIM + c + 3];
  *reinterpret_cast<float4*>(out + (long)tid * 4) = o;
}

// ---------------- host ----------------
extern "C" void kernel_launch(void* const* d_in, const int* in_sizes, int n_in,
                              void* d_out, int out_size, void* d_ws, size_t ws_size,
                              hipStream_t stream) {
  const float* x   = (const float*)d_in[0];
  const float* w   = (const float*)d_in[1];
  const int*   src = (const int*)d_in[2];
  const int*   dst = (const int*)d_in[3];
  const int A = 64;
  const int N = in_sizes[0] / A;
  const int E = in_sizes[2];
  const long Np = (N + 3) & ~3L;   // padded counts keep 16B alignment
  const long Ep = (E + 3) & ~3L;

  // ---- workspace layout ----
  int*   cnt    = (int*)d_ws;                   // Np
  int*   fill   = cnt + Np;                     // Np
  int*   rowptr = fill + Np;                    // Np + 4  (N+1 used)
  int*   eid    = rowptr + Np + 4;              // Ep
  float* ws_agg = (float*)(eid + Ep);           // N*16
  float* gstats = ws_agg + (long)N * 16;        // 7*256
  float* bnab   = gstats + 7 * 256;             // 7*256
  float* wf     = bnab + 7 * 256;               // swizzled weights: 114688 floats max
  float* wfWa   = wf;                           // 64*128
  float* wfWb   = wfWa + 64 * HDIM;             // 4 x 16*128
  float* wfBig  = wfWb + 4 * 16 * HDIM;         // 6 x 128*128
  float* rbuf   = wfBig + 6 * HDIM * HDIM;      // N*128
  float* agg    = rbuf + (long)N * HDIM;        // N*128 (aggregation / 't')
  float* h      = (float*)d_out;                // N*128

  // ---- CSR build + edge pre-aggregation (once; reused by every layer) ----
  k_zeroi<<<1024, 256, 0, stream>>>(cnt, 2 * Np);
  k_zero<<<64, 256, 0, stream>>>(gstats, 7 * 256);
  k_count<<<(E + 255) / 256, 256, 0, stream>>>(dst, cnt, E);
  k_scan<<<1, 1024, 0, stream>>>(cnt, rowptr, N);
  k_fill<<<(E + 255) / 256, 256, 0, stream>>>(dst, rowptr, fill, eid, E);
  k_ws_agg<<<(N + 7) / 8, 256, 0, stream>>>(w, rowptr, eid, ws_agg, N);

  // ---- weight swizzles ----
  const float* Wa  = (const float*)d_in[4];
  k_swz<<<(64 * 32 + 255) / 256, 256, 0, stream>>>(Wa, wfWa, 64);
  k_swz<<<(16 * 32 + 255) / 256, 256, 0, stream>>>((const float*)d_in[6], wfWb, 16);
  for (int i = 0; i < 3; ++i) {
    const int p = 10 + i * 10;
    k_swz<<<(16 * 32 + 255) / 256, 256, 0, stream>>>((const float*)d_in[p + 0],
                                                     wfWb + (1 + i) * 16 * HDIM, 16);
    k_swz<<<(128 * 32 + 255) / 256, 256, 0, stream>>>((const float*)d_in[p + 2],
                                                      wfBig + (2 * i + 0) * HDIM * HDIM, 128);
    k_swz<<<(128 * 32 + 255) / 256, 256, 0, stream>>>((const float*)d_in[p + 4],
                                                      wfBig + (2 * i + 1) * HDIM * HDIM, 128);
  }

  const int gemm_blocks = (N + 127) / 128;
  const int agg_blocks = (N + 7) / 8;
  const int app_blocks = (N * (HDIM / 4) + 255) / 256;
  const float invN = 1.0f / (float)N;

  // ---- layer 0: h = BN(relu(x@Wa + ba + ws_agg@Wb0 + deg*bb0))
  {
    const float* ba  = (const float*)d_in[5];
    const float* bb0 = (const float*)d_in[7];
    const float* g0  = (const float*)d_in[8];
    const float* be0 = (const float*)d_in[9];
    k_gemm<<<gemm_blocks, 256, 0, stream>>>(x, A, wfWa, ba, ws_agg, wfWb, bb0, rowptr,
                                            nullptr, rbuf, gstats, N);
    k_bn_fin<<<1, HDIM, 0, stream>>>(gstats, g0, be0, bnab, invN);
    k_bn_apply<<<app_blocks, 256, 0, stream>>>(rbuf, bnab, h, N);
  }

  // ---- layers 1..3 ----
  for (int i = 0; i < 3; ++i) {
    const int p = 10 + i * 10;
    const float* bb  = (const float*)d_in[p + 1];
    const float* b1  = (const float*)d_in[p + 3];
    const float* b2  = (const float*)d_in[p + 5];
    const float* g1  = (const float*)d_in[p + 6];
    const float* be1 = (const float*)d_in[p + 7];
    const float* g2  = (const float*)d_in[p + 8];
    const float* be2 = (const float*)d_in[p + 9];
    float* st1 = gstats + (1 + 2 * i) * 256;
    float* st2 = gstats + (2 + 2 * i) * 256;
    float* ab1 = bnab + (1 + 2 * i) * 256;
    float* ab2 = bnab + (2 + 2 * i) * 256;
    float* wf1 = wfBig + (2 * i + 0) * HDIM * HDIM;
    float* wf2 = wfBig + (2 * i + 1) * HDIM * HDIM;
    float* wfb = wfWb + (1 + i) * 16 * HDIM;

    // agg = sum over incoming edges of h[src]  (pure gather, no zero-fill needed)
    k_aggregate<<<agg_blocks, 256, 0, stream>>>(h, rowptr, eid, src, agg, N);
    // t = BN(relu(agg@W1 + b1 + ws_agg@Wb + deg*bb))
    k_gemm<<<gemm_blocks, 256, 0, stream>>>(agg, HDIM, wf1, b1, ws_agg, wfb, bb, rowptr,
                                            nullptr, rbuf, st1, N);
    k_bn_fin<<<1, HDIM, 0, stream>>>(st1, g1, be1, ab1, invN);
    k_bn_apply<<<app_blocks, 256, 0, stream>>>(rbuf, ab1, agg /* t overwrites agg */, N);
    // h = BN(relu(t@W2 + b2 + h))
    k_gemm<<<gemm_blocks, 256, 0, stream>>>(agg, HDIM, wf2, b2, nullptr, nullptr, nullptr,
                                            rowptr, h /* residual */, rbuf, st2, N);
    k_bn_fin<<<1, HDIM, 0, stream>>>(st2, g2, be2, ab2, invN);
    k_bn_apply<<<app_blocks, 256, 0, stream>>>(rbuf, ab2, h, N);
  }
}